// BadBatchAllLoss_83915071030173
// MI455X (gfx1250) — compile-verified
//
#include <hip/hip_runtime.h>
#include <hip/hip_bf16.h>

typedef __attribute__((ext_vector_type(2))) float v2f;
typedef __attribute__((ext_vector_type(8))) float v8f;

#define NPTS 512
#define DIM  128
#define EPSV 1e-6f

// ---------------------------------------------------------------------------
// Kernel A: per-row stats  r[i] = ||x_i||^2 , s[i] = sum_d x_i[d]
// ---------------------------------------------------------------------------
__global__ __launch_bounds__(128) void row_stats_kernel(
    const float* __restrict__ X, float* __restrict__ r, float* __restrict__ s)
{
    __shared__ float sh_r[128];
    __shared__ float sh_s[128];
    const int row = blockIdx.x;
    const int t   = threadIdx.x;
    float v = X[row * DIM + t];
    sh_r[t] = v * v;
    sh_s[t] = v;
    __syncthreads();
    for (int off = 64; off > 0; off >>= 1) {
        if (t < off) { sh_r[t] += sh_r[t + off]; sh_s[t] += sh_s[t + off]; }
        __syncthreads();
    }
    if (t == 0) { r[row] = sh_r[0]; s[row] = sh_s[0]; }
}

// ---------------------------------------------------------------------------
// Kernel B: Gram via V_WMMA_F32_16X16X4_F32, one wave per 16x16 tile.
// dist[i,j] = sqrt(max(0, r_i + r_j - 2*G_ij + 2*eps*(s_i - s_j) + D*eps^2))
//
// A (16x4 f32): lane L holds A[L%16][v + 2*(L/16)] in VGPR v.
// B (4x16 f32): lane L holds B[v + 2*(L/16)][L%16].  For B = X^T this is the
// SAME per-lane load pattern as A (a float2 from row-major X).
// C/D (16x16 f32): VGPR v, lane L -> (M = v + 8*(L/16), N = L%16).
// ---------------------------------------------------------------------------
__global__ __launch_bounds__(256) void gram_dist_kernel(
    const float* __restrict__ X, const float* __restrict__ r,
    const float* __restrict__ s, float* __restrict__ dist)
{
    const int lane = threadIdx.x & 31;
    const int wave = threadIdx.x >> 5;
    const int tile = blockIdx.x * 8 + wave;      // 1024 tiles (32x32)
    const int ti = tile >> 5;                    // row-tile
    const int tj = tile & 31;                    // col-tile
    const int m     = lane & 15;
    const int hi    = lane >> 4;                 // 0 or 1 (lane half)
    const int khalf = hi << 1;                   // K sub-offset 0 or 2

    const float* rowA = X + (ti * 16 + m) * DIM + khalf;
    const float* rowB = X + (tj * 16 + m) * DIM + khalf;

    v8f acc = {};
#pragma unroll
    for (int k0 = 0; k0 < DIM; k0 += 4) {
        v2f a = *(const v2f*)(rowA + k0);
        v2f b = *(const v2f*)(rowB + k0);
        // (neg_a, A, neg_b, B, c_mod, C, reuse_a, reuse_b)
        acc = __builtin_amdgcn_wmma_f32_16x16x4_f32(
            false, a, false, b, (short)0, acc, false, false);
    }

    const int   col = tj * 16 + m;
    const float rj  = r[col];
    const float sj  = s[col];
#pragma unroll
    for (int v = 0; v < 8; ++v) {
        const int row = ti * 16 + v + 8 * hi;
        float d2 = r[row] + rj - 2.0f * acc[v]
                 + 2.0f * EPSV * (s[row] - sj)
                 + (float)DIM * EPSV * EPSV;
        dist[row * NPTS + col] = sqrtf(fmaxf(d2, 0.0f));
    }
}

// ---------------------------------------------------------------------------
// Kernel C: per-anchor triplet scan. Block i keeps dist-row i + labels in LDS.
// sum_i = sum over valid (j,g): relu(d_ij - d_ig + margin)
// cnt_i = pos_i * neg_i   (analytic valid-triple count for anchor i)
// Deterministic fixed-order LDS tree reduction.
// ---------------------------------------------------------------------------
__global__ __launch_bounds__(256) void loss_kernel(
    const float* __restrict__ dist, const int* __restrict__ labels,
    const float* __restrict__ marginp,
    float* __restrict__ psum, float* __restrict__ pcnt)
{
    __shared__ float row[NPTS];
    __shared__ int   lab[NPTS];
    __shared__ float red[256];
    __shared__ int   redc[256];

    const int i = blockIdx.x;
    const int t = threadIdx.x;
    const float margin = marginp[0];

    for (int j = t; j < NPTS; j += 256) {
        row[j] = dist[i * NPTS + j];
        lab[j] = labels[j];
    }
    __syncthreads();

    const int labi = lab[i];
    float sum = 0.0f;
    int   same = 0;
    for (int j = t; j < NPTS; j += 256) {
        const bool isSame = (lab[j] == labi);
        if (isSame) same++;
        if (isSame && j != i) {
            const float dij = row[j] + margin;
            for (int g = 0; g < NPTS; ++g) {
                if (lab[g] != labi) {
                    float v = dij - row[g];
                    sum += (v > 0.0f) ? v : 0.0f;
                }
            }
        }
    }
    red[t]  = sum;
    redc[t] = same;
    __syncthreads();
    for (int off = 128; off > 0; off >>= 1) {
        if (t < off) { red[t] += red[t + off]; redc[t] += redc[t + off]; }
        __syncthreads();
    }
    if (t == 0) {
        const int sameTot = redc[0];           // includes self
        const int pos = sameTot - 1;
        const int neg = NPTS - sameTot;
        psum[i] = red[0];
        pcnt[i] = (float)pos * (float)neg;
    }
}

// ---------------------------------------------------------------------------
// Kernel D: final deterministic reduction -> (loss, 0, 0, 0)
// ---------------------------------------------------------------------------
__global__ __launch_bounds__(512) void finalize_kernel(
    const float* __restrict__ psum, const float* __restrict__ pcnt,
    float* __restrict__ out)
{
    __shared__ float a[512];
    __shared__ float b[512];
    const int t = threadIdx.x;
    a[t] = psum[t];
    b[t] = pcnt[t];
    __syncthreads();
    for (int off = 256; off > 0; off >>= 1) {
        if (t < off) { a[t] += a[t + off]; b[t] += b[t + off]; }
        __syncthreads();
    }
    if (t == 0) {
        out[0] = a[0] / b[0];
        out[1] = 0.0f;
        out[2] = 0.0f;
        out[3] = 0.0f;
    }
}

extern "C" void kernel_launch(void* const* d_in, const int* in_sizes, int n_in,
                              void* d_out, int out_size, void* d_ws, size_t ws_size,
                              hipStream_t stream) {
    (void)in_sizes; (void)n_in; (void)out_size; (void)ws_size;
    const float* X      = (const float*)d_in[0];   // [512,128] fp32
    const int*   labels = (const int*)  d_in[1];   // [512] int32
    const float* margin = (const float*)d_in[2];   // scalar

    float* ws   = (float*)d_ws;
    float* r    = ws;                        // 512
    float* s    = ws + NPTS;                 // 512
    float* dist = ws + 2 * NPTS;             // 512*512
    float* psum = dist + NPTS * NPTS;        // 512
    float* pcnt = psum + NPTS;               // 512

    row_stats_kernel<<<NPTS, 128, 0, stream>>>(X, r, s);
    gram_dist_kernel<<<128, 256, 0, stream>>>(X, r, s, dist);
    loss_kernel<<<NPTS, 256, 0, stream>>>(dist, labels, margin, psum, pcnt);
    finalize_kernel<<<1, 512, 0, stream>>>(psum, pcnt, (float*)d_out);
}